// MyModel_61933428413289
// MI455X (gfx1250) — compile-verified
//
#include <hip/hip_runtime.h>

typedef __attribute__((ext_vector_type(2))) float v2f;
typedef __attribute__((ext_vector_type(8))) float v8f;

#define NDIM 10
#define TILES_PER_WAVE 4
#define WAVES_PER_BLOCK 8

// ---------------------------------------------------------------------------
// Kernel 1: invert the 10x10 A (Gauss-Jordan, partial pivoting), store
// B-matrix for WMMA:  Bp[k*16+n] = invA[n][k], zero-padded to 16x16.
// out[m][n] = sum_k X[m][k] * Bp[k][n] = (A^-1 x_m)[n]
// ---------------------------------------------------------------------------
__global__ void invert10_kernel(const float* __restrict__ A, float* __restrict__ Bp) {
    if (threadIdx.x != 0) return;
    float M[NDIM][NDIM];
    float Iv[NDIM][NDIM];
    for (int i = 0; i < NDIM; ++i)
        for (int j = 0; j < NDIM; ++j) {
            M[i][j]  = A[i * NDIM + j];
            Iv[i][j] = (i == j) ? 1.0f : 0.0f;
        }
    for (int c = 0; c < NDIM; ++c) {
        int p = c;
        float best = fabsf(M[c][c]);
        for (int r = c + 1; r < NDIM; ++r) {
            float v = fabsf(M[r][c]);
            if (v > best) { best = v; p = r; }
        }
        if (p != c) {
            for (int j = 0; j < NDIM; ++j) {
                float t = M[c][j];  M[c][j]  = M[p][j];  M[p][j]  = t;
                t       = Iv[c][j]; Iv[c][j] = Iv[p][j]; Iv[p][j] = t;
            }
        }
        float inv = 1.0f / M[c][c];
        for (int j = 0; j < NDIM; ++j) { M[c][j] *= inv; Iv[c][j] *= inv; }
        for (int r = 0; r < NDIM; ++r) {
            if (r == c) continue;
            float f = M[r][c];
            for (int j = 0; j < NDIM; ++j) {
                M[r][j]  -= f * M[c][j];
                Iv[r][j] -= f * Iv[c][j];
            }
        }
    }
    for (int i = 0; i < 256; ++i) Bp[i] = 0.0f;
    for (int k = 0; k < NDIM; ++k)
        for (int n = 0; n < NDIM; ++n)
            Bp[k * 16 + n] = Iv[n][k];
}

// ---------------------------------------------------------------------------
// Kernel 2: out[b,:] = A^-1 x[b,:] via V_WMMA_F32_16X16X4_F32.
// Each wave: 4 consecutive 16-row tiles (64 rows, 2560 contiguous bytes).
//   Phase 1: 12 unconditional global_load_b64 (clamped addr + cndmask zero)
//   Phase 2: 12 back-to-back WMMAs (EXEC all-1s, no branches before them)
//   Phase 3: scatter D-frags to LDS [m*10+n], barrier,
//            20 fully-coalesced 128B nontemporal output stores per wave.
//
// A-frag (32-bit 16x4): lanes 0-15 -> M=lane,    K={k0,k0+1};
//                       lanes 16-31 -> M=lane-16, K={k0+2,k0+3}.
// D (32-bit 16x16):     vgpr r, lanes 0-15 -> M=r,   N=lane;
//                               lanes 16-31 -> M=r+8, N=lane-16.
// ---------------------------------------------------------------------------
__global__ void __launch_bounds__(256) solve_kernel(const float* __restrict__ x,
                                                    const float* __restrict__ Bp,
                                                    float* __restrict__ out,
                                                    int nrows) {
    __shared__ float lds[WAVES_PER_BLOCK * TILES_PER_WAVE * 160];

    const int lane   = threadIdx.x & 31;
    const int waveId = threadIdx.x >> 5;
    const int n      = lane & 15;           // tile column (B/D) / tile row (A)
    const int khalf  = (lane >> 4) * 2;     // 0 or 2: which K-pair this lane holds
    const bool loK   = (khalf == 0);

    const long waveTile0 = ((long)blockIdx.x * WAVES_PER_BLOCK + waveId) * TILES_PER_WAVE;
    const long rowBase0  = waveTile0 * 16;  // first row this wave handles
    const long lastRow   = (long)nrows - 1;

    // --- B-fragments: loaded once per wave, shared by all 4 tiles ----------
    v2f bf[3];
#pragma unroll
    for (int kk = 0; kk < 3; ++kk) {
        const int k0 = kk * 4;
        bf[kk].x = Bp[(k0 + khalf + 0) * 16 + n];
        bf[kk].y = Bp[(k0 + khalf + 1) * 16 + n];
    }

    // --- Phase 1: issue all A-fragment loads (no EXEC divergence) ----------
    v2f a[TILES_PER_WAVE][3];
#pragma unroll
    for (int t = 0; t < TILES_PER_WAVE; ++t) {
        const long myRow   = rowBase0 + t * 16 + n;
        const bool rowOk   = myRow < (long)nrows;
        const long rowSafe = rowOk ? myRow : lastRow;      // clamp, stay in-bounds
        const float* rowPtr = x + rowSafe * NDIM;
        // kb = k0 + khalf: {0,2}, {4,6}, {8,10}; kb==10 has no data -> load
        // the safe kb=8 address for those lanes and select zero afterwards.
        v2f a0 = *reinterpret_cast<const v2f*>(rowPtr + 0 + khalf);
        v2f a1 = *reinterpret_cast<const v2f*>(rowPtr + 4 + khalf);
        v2f a2 = *reinterpret_cast<const v2f*>(rowPtr + 8);
        const v2f z = (v2f){0.0f, 0.0f};
        a[t][0] = rowOk ? a0 : z;
        a[t][1] = rowOk ? a1 : z;
        a[t][2] = (rowOk && loK) ? a2 : z;
    }

    // --- Phase 2: 12 WMMAs, EXEC all-1s ------------------------------------
    v8f acc[TILES_PER_WAVE];
#pragma unroll
    for (int t = 0; t < TILES_PER_WAVE; ++t) {
        v8f c = {0.0f, 0.0f, 0.0f, 0.0f, 0.0f, 0.0f, 0.0f, 0.0f};
        c = __builtin_amdgcn_wmma_f32_16x16x4_f32(false, a[t][0], false, bf[0],
                                                  (short)0, c, false, false);
        c = __builtin_amdgcn_wmma_f32_16x16x4_f32(false, a[t][1], false, bf[1],
                                                  (short)0, c, false, false);
        c = __builtin_amdgcn_wmma_f32_16x16x4_f32(false, a[t][2], false, bf[2],
                                                  (short)0, c, false, false);
        acc[t] = c;
    }

    // --- Phase 3: scatter D-frags into per-wave LDS region -----------------
    float* myLds = lds + waveId * (TILES_PER_WAVE * 160);
    const int mBase = (lane >> 4) * 8;
    if (n < NDIM) {
#pragma unroll
        for (int t = 0; t < TILES_PER_WAVE; ++t) {
            float* tileLds = myLds + t * 160;
#pragma unroll
            for (int r = 0; r < 8; ++r) {
                tileLds[(mBase + r) * NDIM + n] = acc[t][r];
            }
        }
    }

    __syncthreads();

    // --- coalesced nontemporal write-out: 640 dwords per wave --------------
    const long outBase = rowBase0 * NDIM;
    const long outLim  = (long)nrows * NDIM;
#pragma unroll
    for (int i = 0; i < 4 * 5; ++i) {
        const long idx = outBase + lane + 32 * i;
        if (idx < outLim) {
            __builtin_nontemporal_store(myLds[lane + 32 * i], &out[idx]);
        }
    }
}

// ---------------------------------------------------------------------------
extern "C" void kernel_launch(void* const* d_in, const int* in_sizes, int n_in,
                              void* d_out, int out_size, void* d_ws, size_t ws_size,
                              hipStream_t stream) {
    (void)n_in; (void)out_size; (void)ws_size;
    const float* x = (const float*)d_in[0];   // [B, 10]
    const float* A = (const float*)d_in[1];   // [10, 10]
    float* out = (float*)d_out;               // [B, 10]
    float* Bp  = (float*)d_ws;                // 256 floats of scratch

    const int nrows = in_sizes[0] / NDIM;     // B = 2,000,000

    invert10_kernel<<<1, 32, 0, stream>>>(A, Bp);

    const long tiles  = ((long)nrows + 15) / 16;
    const long waves  = (tiles + TILES_PER_WAVE - 1) / TILES_PER_WAVE;
    const int  blocks = (int)((waves + WAVES_PER_BLOCK - 1) / WAVES_PER_BLOCK);
    solve_kernel<<<blocks, 256, 0, stream>>>(x, Bp, out, nrows);
}